// GraphCoordinator_7705171329735
// MI455X (gfx1250) — compile-verified
//
#include <hip/hip_runtime.h>

// ---------------------------------------------------------------------------
// GraphCoordinator exact-row-match replace, MI455X / gfx1250.
//
// out[n, :] = running-replace of x[n, :]:
//   for p in 0..15: if row == last_updated_param[p] (IEEE float eq, all 128
//   elems) then row = learnable_param[p]   (later p sees the updated row)
//
// Pure HBM stream: 512 MB read + 512 MB write -> ~44 us floor at 23.3 TB/s.
// One wave32 per row (lane = 16B chunk, 32*16B = 512B row), non-temporal
// b128 load/store, params staged to LDS with gfx1250 async-to-LDS.
// Per-row match filter = ONE v_cmp_eq_f32: broadcast row element 0 (SGPR via
// v_readfirstlane) against a lane-resident vector of the 16 params' element 0;
// ballot != 0 -> rare slow path does the exact sequential full compares.
// ---------------------------------------------------------------------------

typedef float v4f __attribute__((ext_vector_type(4)));
typedef int   v4i __attribute__((ext_vector_type(4)));

// gfx1250 async-LDS builtin pointer types: (int4 AS1*, int4 AS3*, imm, imm)
typedef __attribute__((address_space(1))) v4i gas_v4i;
typedef __attribute__((address_space(3))) v4i las_v4i;

#ifndef __has_builtin
#define __has_builtin(x) 0
#endif

#if __has_builtin(__builtin_amdgcn_global_load_async_to_lds_b128) && \
    __has_builtin(__builtin_amdgcn_s_wait_asynccnt)
#define USE_ASYNC_LDS 1
#else
#define USE_ASYNC_LDS 0
#endif

static constexpr int F     = 128;  // node feature dim
static constexpr int P     = 16;   // num coordinator params
static constexpr int BLOCK = 256;  // 8 waves (wave32)
static constexpr int WAVES = BLOCK / 32;

__device__ __forceinline__ unsigned wave_ballot32(bool m) {
#if __has_builtin(__builtin_amdgcn_ballot_w32)
  return __builtin_amdgcn_ballot_w32(m);
#else
  return (unsigned)__ballot((int)m);
#endif
}

__global__ __launch_bounds__(BLOCK) void coord_replace_kernel(
    const float* __restrict__ x,
    const float* __restrict__ learn,
    const float* __restrict__ last,
    float* __restrict__ out,
    int n_rows) {
  __shared__ float lds_last[P * F];   // 8 KB
  __shared__ float lds_learn[P * F];  // 8 KB

  // ---- stage both param tables into LDS (16 KB) via async-to-LDS ---------
#if USE_ASYNC_LDS
  for (int i = threadIdx.x; i < (P * F) / 4; i += BLOCK) {
    __builtin_amdgcn_global_load_async_to_lds_b128(
        (gas_v4i*)(unsigned long long)(last + i * 4),
        (las_v4i*)(unsigned)(unsigned long long)(lds_last + i * 4),
        0, 0);
    __builtin_amdgcn_global_load_async_to_lds_b128(
        (gas_v4i*)(unsigned long long)(learn + i * 4),
        (las_v4i*)(unsigned)(unsigned long long)(lds_learn + i * 4),
        0, 0);
  }
  __builtin_amdgcn_s_wait_asynccnt(0);
#else
  for (int i = threadIdx.x; i < (P * F) / 4; i += BLOCK) {
    ((v4f*)lds_last)[i]  = ((const v4f*)last)[i];
    ((v4f*)lds_learn)[i] = ((const v4f*)learn)[i];
  }
#endif
  __syncthreads();

  const int lane = threadIdx.x & 31;
  const int wid  = threadIdx.x >> 5;

  // Lane-resident filter: lane l holds element 0 of last_param[l & 15].
  // (lanes 16..31 duplicate lanes 0..15; extra filter hits are harmless)
  const float filtf = lds_last[(lane & 15) * F];

  const v4f* lastv  = (const v4f*)lds_last;
  const v4f* learnv = (const v4f*)lds_learn;

  const long long row0  = (long long)blockIdx.x * WAVES + wid;
  const long long rstep = (long long)gridDim.x * WAVES;

  // 32-bit byte offsets: N*F*4 = 512 MB < 4 GB, so SADDR + 32-bit voffset
  // (GVS addressing) works and costs one v_add per iteration.
  const char* xb = (const char*)x;
  char*       ob = (char*)out;
  unsigned off        = (unsigned)row0 * (unsigned)(F * 4) + (unsigned)lane * 16u;
  const unsigned step = (unsigned)rstep * (unsigned)(F * 4);

  for (long long row = row0; row < (long long)n_rows; row += rstep, off += step) {
    // speculative prefetch of this wave's next grid-stride row
    __builtin_prefetch((const void*)(xb + off + step), 0, 0);

    // whole 512B row in one wave: lane holds cols [4*lane, 4*lane+3]
    v4f cur = __builtin_nontemporal_load((const v4f*)(xb + off));

    // --- single-VALU filter: row elem 0 vs all 16 param elem-0 values -----
    // float '==' matches reference semantics (+0==-0; NaN rows never match,
    // so a NaN filter-reject is exact).
    const float r0 = __uint_as_float((unsigned)__builtin_amdgcn_readfirstlane(
        (int)__float_as_uint(cur.x)));
    const unsigned hit = wave_ballot32(r0 == filtf);

    if (hit != 0u) {  // rare (~16 rows in 1M): uniform branch, EXEC stays full
#pragma unroll 1
      for (int p = 0; p < P; ++p) {
        const v4f lp = lastv[p * (F / 4) + lane];
        const bool m = (cur.x == lp.x) & (cur.y == lp.y) &
                       (cur.z == lp.z) & (cur.w == lp.w);
        if (wave_ballot32(m) == 0xffffffffu) {
          cur = learnv[p * (F / 4) + lane];  // later p compares updated row
        }
      }
    }

    __builtin_nontemporal_store(cur, (v4f*)(ob + off));
  }
}

extern "C" void kernel_launch(void* const* d_in, const int* in_sizes, int n_in,
                              void* d_out, int out_size, void* d_ws,
                              size_t ws_size, hipStream_t stream) {
  (void)n_in; (void)out_size; (void)d_ws; (void)ws_size;
  // setup_inputs() order: x, batch, learnable_param, last_updated_param
  const float* x     = (const float*)d_in[0];
  const float* learn = (const float*)d_in[2];
  const float* last  = (const float*)d_in[3];
  float* out         = (float*)d_out;

  const int n_rows = in_sizes[0] / F;  // N = 1,000,000

  int blocks = (n_rows + WAVES - 1) / WAVES;
  if (blocks > 8192) blocks = 8192;  // grid-stride; bounds param L2 re-reads
  if (blocks < 1) blocks = 1;

  coord_replace_kernel<<<blocks, BLOCK, 0, stream>>>(x, learn, last, out,
                                                     n_rows);
}